// Attention_Based_Encoder_20856361189563
// MI455X (gfx1250) — compile-verified
//
#include <hip/hip_runtime.h>
#include <hip/hip_bf16.h>
#include <math.h>

// Problem constants (match reference setup_inputs)
#define BB 32
#define MM 2048
#define DD 1024
#define CD 5120   // C*D
#define JC 16     // j-chunks for enc partial sums (chunk = 128)

typedef float v2f __attribute__((ext_vector_type(2)));
typedef float v8f __attribute__((ext_vector_type(8)));

// ---------------------------------------------------------------------------
// K1: a[b,d] = sum_k P[d,k] * y[b,k]   via V_WMMA_F32_16X16X4_F32
// One wave (32 threads) per 16x16 output tile. A = y tile (16x4, M=b rows),
// B = P^T tile (4x16, N=d cols). K loop over 5120 in steps of 4.
// Fragment layouts per CDNA5 ISA 7.12.2:
//   A 16x4 f32: elem v of lane L -> row M=L%16, K = v + 2*(L>=16)
//   B 4x16 f32: elem v of lane L -> K = v + 2*(L>=16), col N=L%16
//   C 16x16 f32: elem v of lane L -> row M = v + 8*(L>=16), col N=L%16
// ---------------------------------------------------------------------------
__global__ __launch_bounds__(32) void a_gemm_wmma_kernel(
    const float* __restrict__ y,   // [B, CD]
    const float* __restrict__ P,   // [D, CD]
    float* __restrict__ a)         // [B, D]
{
  const int b0 = blockIdx.x * 16;          // 2 tiles over B
  const int d0 = blockIdx.y * 16;          // 64 tiles over D
  const int lane = threadIdx.x;
  const int half = lane >> 4;              // 0 / 1
  const int l16  = lane & 15;

  const float* yrow = y + (size_t)(b0 + l16) * CD + 2 * half;   // A row
  const float* prow = P + (size_t)(d0 + l16) * CD + 2 * half;   // B col (P^T row pair)

  v8f acc = {};
  for (int k = 0; k < CD; k += 4) {
    v2f af, bf;
    af.x = yrow[k];  af.y = yrow[k + 1];
    bf.x = prow[k];  bf.y = prow[k + 1];
    acc = __builtin_amdgcn_wmma_f32_16x16x4_f32(
        /*neg_a=*/false, af, /*neg_b=*/false, bf,
        /*c_mod=*/(short)0, acc, /*reuse_a=*/false, /*reuse_b=*/false);
  }

  #pragma unroll
  for (int v = 0; v < 8; ++v) {
    const int mr = v + 8 * half;
    a[(size_t)(b0 + mr) * DD + d0 + l16] = acc[v];
  }
}

// ---------------------------------------------------------------------------
// K2: logits[b,m] = dot(x[b,m,:], a[b,:]).  256 threads = 8 waves, one wave
// per (b,m) row; a[b] staged in LDS; float4 (b128) loads; wave32 shuffle
// reduction.
// ---------------------------------------------------------------------------
__global__ __launch_bounds__(256) void logits_kernel(
    const float* __restrict__ x,      // [B, M, D]
    const float* __restrict__ a,      // [B, D]
    float* __restrict__ logits)       // [B, M]
{
  __shared__ float sa[DD];
  const int row0 = blockIdx.x * 8;          // 8 rows per block, same b
  const int b = row0 >> 11;                 // /MM
  for (int i = threadIdx.x; i < DD; i += 256) sa[i] = a[(size_t)b * DD + i];
  __syncthreads();

  const int wid  = threadIdx.x >> 5;
  const int lane = threadIdx.x & 31;
  const int row  = row0 + wid;

  const float4* xr = (const float4*)(x + (size_t)row * DD);
  const float4* ar = (const float4*)sa;
  float acc = 0.f;
  #pragma unroll
  for (int i = 0; i < DD / 4 / 32; ++i) {
    float4 xv = xr[lane + 32 * i];
    float4 av = ar[lane + 32 * i];
    acc = fmaf(xv.x, av.x, acc);
    acc = fmaf(xv.y, av.y, acc);
    acc = fmaf(xv.z, av.z, acc);
    acc = fmaf(xv.w, av.w, acc);
  }
  #pragma unroll
  for (int off = 16; off; off >>= 1) acc += __shfl_xor(acc, off, 32);
  if (lane == 0) logits[row] = acc;
}

// ---------------------------------------------------------------------------
// K3: per-batch softmax over M, then fold window weights:
//   w[j] = 0.5 * sum_{m=max(j-1,0)}^{min(j+2,M-1)} p[m]  for j < M-1, w[M-1]=0
// One block per b.
// ---------------------------------------------------------------------------
__global__ __launch_bounds__(256) void softmax_w_kernel(
    const float* __restrict__ logits,  // [B, M]
    float* __restrict__ w)             // [B, M]
{
  __shared__ float sp[MM];
  __shared__ float redmax[8];
  __shared__ float redsum[8];

  const int b = blockIdx.x;
  const int t = threadIdx.x;
  const int wid = t >> 5, lane = t & 31;
  const float* lg = logits + (size_t)b * MM;

  float lmax = -INFINITY;
  for (int i = t; i < MM; i += 256) {
    float v = lg[i];
    sp[i] = v;
    lmax = fmaxf(lmax, v);
  }
  #pragma unroll
  for (int off = 16; off; off >>= 1) lmax = fmaxf(lmax, __shfl_xor(lmax, off, 32));
  if (lane == 0) redmax[wid] = lmax;
  __syncthreads();
  float bmax = redmax[0];
  #pragma unroll
  for (int i = 1; i < 8; ++i) bmax = fmaxf(bmax, redmax[i]);

  float lsum = 0.f;
  for (int i = t; i < MM; i += 256) {
    float e = __expf(sp[i] - bmax);
    sp[i] = e;
    lsum += e;
  }
  #pragma unroll
  for (int off = 16; off; off >>= 1) lsum += __shfl_xor(lsum, off, 32);
  if (lane == 0) redsum[wid] = lsum;
  __syncthreads();
  float bsum = 0.f;
  #pragma unroll
  for (int i = 0; i < 8; ++i) bsum += redsum[i];
  const float inv = 0.5f / bsum;   // fold softmax norm and the 1/Q=0.5

  for (int j = t; j < MM; j += 256) {
    float acc = 0.f;
    if (j < MM - 1) {
      const int lo = (j - 1 < 0) ? 0 : j - 1;
      const int hi = (j + 2 > MM - 1) ? MM - 1 : j + 2;
      for (int m = lo; m <= hi; ++m) acc += sp[m];
      acc *= inv;
    }
    w[(size_t)b * MM + j] = acc;
  }
}

// ---------------------------------------------------------------------------
// K4: partial[jc][b][d] = sum_{j in chunk jc} x[b,j,d] * w[b,j]
// grid (B, JC); each thread owns 4 consecutive d (float4 / b128 loads), one
// block covers the full D row; w chunk staged in LDS; two independent float4
// accumulators for ILP.
// ---------------------------------------------------------------------------
__global__ __launch_bounds__(256) void enc_partial_kernel(
    const float* __restrict__ x,       // [B, M, D]
    const float* __restrict__ w,       // [B, M]
    float* __restrict__ partial)       // [JC, B, D]
{
  __shared__ float sw[MM / JC];        // 128
  const int b  = blockIdx.x;
  const int jc = blockIdx.y;
  const int j0 = jc * (MM / JC);
  const int t  = threadIdx.x;

  if (t < MM / JC) sw[t] = w[(size_t)b * MM + j0 + t];
  __syncthreads();

  const float4* xp = (const float4*)(x + ((size_t)b * MM + j0) * DD) + t;
  float4 acc0 = {0.f, 0.f, 0.f, 0.f};
  float4 acc1 = {0.f, 0.f, 0.f, 0.f};
  #pragma unroll 4
  for (int j = 0; j < MM / JC; j += 2) {
    float4 xv0 = xp[(size_t)(j + 0) * (DD / 4)];
    float4 xv1 = xp[(size_t)(j + 1) * (DD / 4)];
    const float w0 = sw[j + 0];
    const float w1 = sw[j + 1];
    acc0.x = fmaf(xv0.x, w0, acc0.x);
    acc0.y = fmaf(xv0.y, w0, acc0.y);
    acc0.z = fmaf(xv0.z, w0, acc0.z);
    acc0.w = fmaf(xv0.w, w0, acc0.w);
    acc1.x = fmaf(xv1.x, w1, acc1.x);
    acc1.y = fmaf(xv1.y, w1, acc1.y);
    acc1.z = fmaf(xv1.z, w1, acc1.z);
    acc1.w = fmaf(xv1.w, w1, acc1.w);
  }
  float4 r;
  r.x = acc0.x + acc1.x;
  r.y = acc0.y + acc1.y;
  r.z = acc0.z + acc1.z;
  r.w = acc0.w + acc1.w;
  ((float4*)(partial + ((size_t)jc * BB + b) * DD))[t] = r;
}

// ---------------------------------------------------------------------------
// K5: deterministic reduction of partials -> enc[b,d]
// ---------------------------------------------------------------------------
__global__ __launch_bounds__(256) void enc_reduce_kernel(
    const float* __restrict__ partial,  // [JC, B, D]
    float* __restrict__ out)            // [B, D]
{
  const int idx = blockIdx.x * 256 + threadIdx.x;   // over B*D
  float acc = 0.f;
  #pragma unroll
  for (int jc = 0; jc < JC; ++jc) acc += partial[(size_t)jc * (BB * DD) + idx];
  out[idx] = acc;
}

// ---------------------------------------------------------------------------
extern "C" void kernel_launch(void* const* d_in, const int* in_sizes, int n_in,
                              void* d_out, int out_size, void* d_ws, size_t ws_size,
                              hipStream_t stream) {
  const float* x = (const float*)d_in[0];   // [B, M, D]
  const float* y = (const float*)d_in[1];   // [B, C*D]
  const float* P = (const float*)d_in[2];   // [D, C*D]
  float* out = (float*)d_out;               // [B, D]

  // workspace layout (floats)
  float* ws      = (float*)d_ws;
  float* ws_a    = ws;                               // B*D      = 32768
  float* ws_lg   = ws_a  + (size_t)BB * DD;          // B*M      = 65536
  float* ws_w    = ws_lg + (size_t)BB * MM;          // B*M      = 65536
  float* ws_part = ws_w  + (size_t)BB * MM;          // JC*B*D   = 524288

  // K1: a = y @ P^T  (WMMA f32 16x16x4)
  a_gemm_wmma_kernel<<<dim3(BB / 16, DD / 16), 32, 0, stream>>>(y, P, ws_a);
  // K2: logits (first 268 MB streaming pass over x, b128 loads)
  logits_kernel<<<(BB * MM) / 8, 256, 0, stream>>>(x, ws_a, ws_lg);
  // K3: softmax + window-weight fold
  softmax_w_kernel<<<BB, 256, 0, stream>>>(ws_lg, ws_w);
  // K4: chunked weighted sum over sequence (second 268 MB pass, b128 loads)
  enc_partial_kernel<<<dim3(BB, JC), 256, 0, stream>>>(x, ws_w, ws_part);
  // K5: final reduce
  enc_reduce_kernel<<<(BB * DD) / 256, 256, 0, stream>>>(ws_part, out);
}